// ODERamp_41094247088632
// MI455X (gfx1250) — compile-verified
//
#include <hip/hip_runtime.h>

// ---------------------------------------------------------------------------
// ODE ramp, collapsed Tsit5.
//   y' = I*(1-y), per-pixel constant I  ==>  one Tsit5 substep is exactly
//   y <- y + (1-y)*g(z), z = I*dt (g = Tsit5 stability polynomial - 1);
//   8 substeps/group collapse to y <- y + (1-y)*G, G = 1-(1-g)^8.
// Memory-bound: 64MiB in + 128MiB out => ~8.6us floor at 23.3 TB/s; no matmul
// exists in this workload, so WMMA would only add shuffle overhead. The CDNA5
// feature that fits is the Tensor Data Mover: each wave32 DMA's its own 8x64
// f32 slab global->LDS with tensor_load_to_lds (TENSORcnt), then consumes only
// its own rows -- no workgroup barrier at all.
// ---------------------------------------------------------------------------

typedef __attribute__((ext_vector_type(4))) unsigned u32x4;
typedef __attribute__((ext_vector_type(8))) unsigned u32x8;

#define BLOCK_THREADS 256
#define TILE_I   16                     // 16x16 output pixels per workgroup
#define TILE_IN  64                     // 64x64 input floats per workgroup
#define ROWS_PER_WAVE 8                 // 64 rows / 8 wave32s

// Tsit5 coefficients (Tsitouras 2011)
#define A21f  0.161f
#define A31f -0.008480655492356989f
#define A32f  0.335480655492357f
#define A41f  2.8971530571054935f
#define A42f -6.359448489975075f
#define A43f  4.3622954328695815f
#define A51f  5.325864828439257f
#define A52f -11.748883564062828f
#define A53f  7.4955393428898365f
#define A54f -0.09249506636175525f
#define A61f  5.86145544294642f
#define A62f -12.92096931784711f
#define A63f  8.159367898576159f
#define A64f -0.071584973281401f
#define A65f -0.028269050394068383f
#define B1f   0.09646076681806523f
#define B2f   0.01f
#define B3f   0.4798896504144996f
#define B4f   1.379008574103742f
#define B5f  -3.290069515436081f
#define B6f   2.324710524099774f

__global__ __launch_bounds__(BLOCK_THREADS)
void oderamp_kernel(const float* __restrict__ in, float* __restrict__ out,
                    int W, int P, int ngroups, float zscale)
{
    __shared__ __align__(16) float tile[TILE_IN * TILE_IN];   // 16 KiB

    const int t  = threadIdx.x;
    const int bx = blockIdx.x;
    const int by = blockIdx.y;
    const int wave = __builtin_amdgcn_readfirstlane(t >> 5);  // wave32 id, uniform
    const int row0 = by * TILE_IN + wave * ROWS_PER_WAVE;     // this wave's slab
    const int col0 = bx * TILE_IN;

    // ---- Tensor Data Mover: DMA this wave's 8x64 f32 slab global -> LDS.
    const unsigned lds0 =
        (unsigned)(size_t)(void*)tile + (unsigned)(wave * ROWS_PER_WAVE * TILE_IN * 4);
    const unsigned long long gaddr =
        (unsigned long long)(size_t)in +
        ((unsigned long long)(unsigned)row0 * (unsigned)W + (unsigned)col0) * 4ull;

    // D# group 0: count=1 | lds_addr | global_addr[56:0] | type=2 ("image")
    u32x4 g0;
    g0.x = 1u;
    g0.y = lds0;
    g0.z = (unsigned)gaddr;
    g0.w = ((unsigned)(gaddr >> 32) & 0x01FFFFFFu) | (2u << 30);

    // D# group 1: wg_mask=0 (not in cluster), data_size=2 (4B), no pad/iterate,
    // tensor_dim0 = tensor_dim1 = W, tile_dim0 = 64, tile_dim1 = 8, tile_dim2=0,
    // tensor_dim0_stride = W (row-major), dim1_stride unused for 2D tiles.
    const unsigned uW = (unsigned)W;
    u32x8 g1;
    g1.s0 = 2u << 16;
    g1.s1 = (uW & 0xFFFFu) << 16;
    g1.s2 = ((uW >> 16) & 0xFFFFu) | ((uW & 0xFFFFu) << 16);
    g1.s3 = ((uW >> 16) & 0xFFFFu) | ((unsigned)TILE_IN << 16);
    g1.s4 = (unsigned)ROWS_PER_WAVE;
    g1.s5 = uW;
    g1.s6 = 0u;
    g1.s7 = 0u;

    // D# groups 2/3: zero (tile_dim2=0 => 2D; higher-dim fields unused)
    u32x4 gz = {0u, 0u, 0u, 0u};

    asm volatile("tensor_load_to_lds %0, %1, %2, %3"
                 :
                 : "s"(g0), "s"(g1), "s"(gz), "s"(gz)
                 : "memory");
    // Wave consumes only the rows it DMA'd itself -> wait own TENSORcnt, no
    // workgroup barrier. Memory clobber keeps ds loads below the wait.
    asm volatile("s_wait_tensorcnt 0x0" ::: "memory");

    // ---- One output pixel per thread: 4x4 box sum from LDS (ds_load_b128).
    const int tx = t & 15;
    const int ty = t >> 4;                       // rows 4*ty..4*ty+3 in [8w, 8w+8)
    const float* bp = &tile[(ty * 4) * TILE_IN + tx * 4];
    float s = 0.0f;
    #pragma unroll
    for (int r = 0; r < 4; ++r) {
        const float4 v = *reinterpret_cast<const float4*>(bp + r * TILE_IN);
        s += (v.x + v.y) + (v.z + v.w);
    }

    // z = I*dt = (blocksum/32768) * (1/(ngroups*8)); zscale exact (2^-23 default)
    const float z = s * zscale;

    // Tsit5 stage multipliers c_i(z): k_i = I*(1-y)*c_i
    const float c2 = 1.0f - z * A21f;
    const float c3 = 1.0f - z * (A31f + A32f * c2);
    const float c4 = 1.0f - z * (A41f + A42f * c2 + A43f * c3);
    const float c5 = 1.0f - z * (A51f + A52f * c2 + A53f * c3 + A54f * c4);
    const float c6 = 1.0f - z * (A61f + A62f * c2 + A63f * c3 + A64f * c4 + A65f * c5);
    const float g  = z * (B1f + B2f * c2 + B3f * c3 + B4f * c4 + B5f * c5 + B6f * c6);
    // one group (8 substeps): G = 1-(1-g)^8 via exact binomial Horner
    const float G = g * (8.0f - g * (28.0f - g * (56.0f - g * (70.0f -
                    g * (56.0f - g * (28.0f - g * (8.0f - g)))))));

    const size_t pix = (size_t)(by * TILE_I + ty) * (size_t)P + (size_t)(bx * TILE_I + tx);
    const size_t gstride = (size_t)P * (size_t)P;

    float y = 0.0f;
    for (int gi = 0; gi < ngroups; ++gi) {
        y = fmaf(1.0f - y, G, y);
        // write-once 128MiB stream: non-temporal store
        __builtin_nontemporal_store(32768.0f * y, out + (size_t)gi * gstride + pix);
    }
}

extern "C" void kernel_launch(void* const* d_in, const int* in_sizes, int n_in,
                              void* d_out, int out_size, void* d_ws, size_t ws_size,
                              hipStream_t stream)
{
    (void)n_in; (void)d_ws; (void)ws_size;
    const float* in = (const float*)d_in[0];
    float* out = (float*)d_out;

    // Input is W x W (power of two; 4096 in the reference harness)
    const long long n0 = (long long)in_sizes[0];
    int W = 1;
    while ((long long)W * (long long)W < n0) W <<= 1;
    const int P = W / 4;                                   // downsampled side
    const int ngroups = (int)((long long)out_size / ((long long)P * (long long)P));

    // z = blocksum * dt / 32768 ; dt = 1/(ngroups*SUBSTEPS), SUBSTEPS = 8
    const float zscale = (float)(1.0 / (32768.0 * (double)ngroups * 8.0));

    dim3 grid(W / TILE_IN, W / TILE_IN);
    dim3 block(BLOCK_THREADS);
    hipLaunchKernelGGL(oderamp_kernel, grid, block, 0, stream,
                       in, out, W, P, ngroups, zscale);
}